// Encoder_86526411145904
// MI455X (gfx1250) — compile-verified
//
#include <hip/hip_runtime.h>

// ---------------------------------------------------------------------------
// 2-layer bipartite hetero-SAGE for MI455X (gfx1250, wave32).
// - Scatter phase: HBM-bound (~2 GB gathered reads @ 23.3 TB/s); b128 gathers
//   + f32 atomics into L2-resident sum buffers.
// - GEMM phase: V_WMMA_F32_16X16X4_F32 (exact f32), weights persistent in
//   VGPRs, A tiles double-buffered into LDS via global_load_async_to_lds_b128
//   (ASYNCcnt + barrier sync). Mean division hoisted out of the hot loop via
//   precomputed 1/deg and post-scaling (row scaling commutes with A@W).
// ---------------------------------------------------------------------------

typedef __attribute__((ext_vector_type(2))) float v2f;
typedef __attribute__((ext_vector_type(8))) float v8f;
typedef int v4i __attribute__((vector_size(16)));   // builtin's pointee type

#define HD   128      // feature dim for every stage
#define LDSS 132      // padded LDS row stride (floats): bank-conflict-free

#if defined(__has_builtin)
#if __has_builtin(__builtin_amdgcn_global_load_async_to_lds_b128)
#define HAVE_ASYNC_LDS 1
#endif
#endif

#define GPTR(p) ((__attribute__((address_space(1))) v4i*)(p))
#define LPTR(p) ((__attribute__((address_space(3))) v4i*)(p))

// ---- degree count (once; same degrees for both layers) --------------------
__global__ void deg_kernel(const int* __restrict__ src, const int* __restrict__ dst,
                           float* __restrict__ cnt_d, float* __restrict__ cnt_g, int E) {
  int stride = gridDim.x * blockDim.x;
  for (int i = blockIdx.x * blockDim.x + threadIdx.x; i < E; i += stride) {
    atomicAdd(&cnt_d[src[i]], 1.0f);
    atomicAdd(&cnt_g[dst[i]], 1.0f);
  }
}

// ---- cnt -> 1/max(cnt,1) in place (one exact divide per node) -------------
__global__ void inv_kernel(float* __restrict__ cnt, int n) {
  int i = blockIdx.x * blockDim.x + threadIdx.x;
  if (i < n) cnt[i] = 1.0f / fmaxf(cnt[i], 1.0f);
}

// ---- edge gather + segment-sum: 32 lanes per edge, float4 per lane --------
__global__ void scatter_sum_kernel(const float* __restrict__ xsrc,
                                   const int* __restrict__ es,
                                   const int* __restrict__ ed,
                                   float* __restrict__ sum, int E) {
  long long stride = (long long)gridDim.x * blockDim.x;
  long long total  = (long long)E * 32;
  for (long long t = (long long)blockIdx.x * blockDim.x + threadIdx.x;
       t < total; t += stride) {
    int e     = (int)(t >> 5);          // wave-uniform: whole wave shares e
    int lane4 = ((int)t & 31) << 2;
    int s = es[e];
    int d = ed[e];
    float4 v = *(const float4*)(xsrc + (long long)s * HD + lane4);  // b128
    float* o = sum + (long long)d * HD + lane4;
    atomicAdd(o + 0, v.x);   // global_atomic_add_f32, L2-resident target
    atomicAdd(o + 1, v.y);
    atomicAdd(o + 2, v.z);
    atomicAdd(o + 3, v.w);
  }
}

// ---- fused SAGE GEMM: out = (sum @ Wl) * inv[row] + bias + xdst @ Wr ------
// 256 threads = 8 waves; each wave owns a 16x16 output tile (16 of 128 cols).
// Weight fragments stay in VGPRs; A tiles double-buffered in LDS via async
// global->LDS copies overlapped with the previous tile's WMMAs.
__global__ __launch_bounds__(256) void sage_gemm_kernel(
    const float* __restrict__ sum,   // [N,128] aggregated sums
    const float* __restrict__ inv,   // [N] 1/max(deg,1)
    const float* __restrict__ xdst,  // [N,128] self features
    const float* __restrict__ Wl,    // [128,128] row-major (K x N)
    const float* __restrict__ Wr,    // [128,128]
    const float* __restrict__ bias,  // [128]
    float* __restrict__ out,         // [N,128]
    int N)
{
  __shared__ float lmean[2][16 * LDSS];
  __shared__ float lself[2][16 * LDSS];
  __shared__ float linv[2][16];

  const int tid   = threadIdx.x;
  const int wave  = tid >> 5;         // 0..7 -> column tile
  const int lane  = tid & 31;
  const int n0    = wave << 4;
  const int ncol  = lane & 15;        // A row index / B,C column index
  const int khalf = (lane >> 4) << 1; // K sub-offset: 0 or 2

  // Persistent B fragments: lane l holds (k, k+1) at column l&15.
  v2f bl[32], br[32];
#pragma unroll
  for (int kk = 0; kk < 32; ++kk) {
    int k = (kk << 2) + khalf;
    bl[kk].x = Wl[(k + 0) * HD + n0 + ncol];
    bl[kk].y = Wl[(k + 1) * HD + n0 + ncol];
    br[kk].x = Wr[(k + 0) * HD + n0 + ncol];
    br[kk].y = Wr[(k + 1) * HD + n0 + ncol];
  }
  const float bv = bias[n0 + ncol];

  const int ntiles = (N + 15) >> 4;

  // Stage one 16x128 sum tile + self tile (+16 inv values) into LDS buf.
  // OOB rows clamp the *address*; their outputs are never stored and rows of
  // A are independent in A@W, so clamping cannot corrupt valid outputs.
  auto stage = [&](int tile, int buf) {
    const int row0 = tile << 4;
#pragma unroll
    for (int p = 0; p < 2; ++p) {
      int idx = (tid + (p << 8)) << 2;  // 0..2044 step 4
      int r   = idx >> 7;
      int c4  = idx & 127;
      int row = row0 + r; if (row > N - 1) row = N - 1;
      const size_t go = (size_t)row * HD + c4;
      const size_t lo = (size_t)r * LDSS + c4;
#if HAVE_ASYNC_LDS
      __builtin_amdgcn_global_load_async_to_lds_b128(
          GPTR(sum + go),  LPTR(&lmean[buf][lo]), 0, 0);
      __builtin_amdgcn_global_load_async_to_lds_b128(
          GPTR(xdst + go), LPTR(&lself[buf][lo]), 0, 0);
#else
      *(float4*)&lmean[buf][lo] = *(const float4*)(sum + go);
      *(float4*)&lself[buf][lo] = *(const float4*)(xdst + go);
#endif
    }
    if (tid < 4) {
      int base = row0 + (tid << 2);
      if (base > N - 4) base = N - 4;   // misplacement hits OOB rows only
#if HAVE_ASYNC_LDS
      __builtin_amdgcn_global_load_async_to_lds_b128(
          GPTR(inv + base), LPTR(&linv[buf][tid << 2]), 0, 0);
#else
      *(float4*)&linv[buf][tid << 2] = *(const float4*)(inv + base);
#endif
    }
  };

  const int t0 = blockIdx.x;
  if (t0 < ntiles) stage(t0, 0);
  int cur = 0;

  for (int tile = t0; tile < ntiles; tile += gridDim.x) {
#if HAVE_ASYNC_LDS
    asm volatile("s_wait_asynccnt 0x0" ::: "memory");
#endif
    __syncthreads();                       // buf `cur` fully staged + visible

    const int nxt = tile + gridDim.x;
    if (nxt < ntiles) stage(nxt, cur ^ 1); // overlap copy with WMMAs below

    v8f cm = {};   // mean path accumulator (scaled by inv at store)
    v8f cs = {};   // self path accumulator
#pragma unroll
    for (int kk = 0; kk < 32; ++kk) {
      int k = (kk << 2) + khalf;
      v2f am = *(const v2f*)&lmean[cur][ncol * LDSS + k];  // ds_load_2addr_b64
      v2f as = *(const v2f*)&lself[cur][ncol * LDSS + k];
      cm = __builtin_amdgcn_wmma_f32_16x16x4_f32(false, am, false, bl[kk],
                                                 (short)0, cm, false, false);
      cs = __builtin_amdgcn_wmma_f32_16x16x4_f32(false, as, false, br[kk],
                                                 (short)0, cs, false, false);
    }

    const int row0 = tile << 4;
#pragma unroll
    for (int r = 0; r < 8; ++r) {
      int m   = ((lane >> 4) << 3) + r;
      int row = row0 + m;
      if (row < N)
        out[(size_t)row * HD + n0 + ncol] = cm[r] * linv[cur][m] + cs[r] + bv;
    }

    __syncthreads();   // all reads of buf `cur` done before it is restaged
    cur ^= 1;
  }
}

static inline int imin(int a, int b) { return a < b ? a : b; }

extern "C" void kernel_launch(void* const* d_in, const int* in_sizes, int n_in,
                              void* d_out, int out_size, void* d_ws, size_t ws_size,
                              hipStream_t stream) {
  const float* x_d   = (const float*)d_in[0];
  const float* x_g   = (const float*)d_in[1];
  const int*   src   = (const int*)d_in[2];
  const int*   dst   = (const int*)d_in[3];
  const float* w1dgl = (const float*)d_in[4];
  const float* b1dg  = (const float*)d_in[5];
  const float* w1dgr = (const float*)d_in[6];
  const float* w1gdl = (const float*)d_in[7];
  const float* b1gd  = (const float*)d_in[8];
  const float* w1gdr = (const float*)d_in[9];
  const float* w2dgl = (const float*)d_in[10];
  const float* b2dg  = (const float*)d_in[11];
  const float* w2dgr = (const float*)d_in[12];
  const float* w2gdl = (const float*)d_in[13];
  const float* b2gd  = (const float*)d_in[14];
  const float* w2gdr = (const float*)d_in[15];

  const int ND = in_sizes[0] / HD;
  const int NG = in_sizes[1] / HD;
  const int E  = in_sizes[2];

  float* ws    = (float*)d_ws;
  float* cnt_d = ws;                       // becomes inv_d after inv_kernel
  float* cnt_g = cnt_d + ND;               // becomes inv_g
  float* sum_g = cnt_g + NG;
  float* sum_d = sum_g + (size_t)NG * HD;
  float* g1    = sum_d + (size_t)ND * HD;
  float* d1    = g1    + (size_t)NG * HD;

  float* d2 = (float*)d_out;               // outputs concatenated: (d2, g2)
  float* g2 = d2 + (size_t)ND * HD;

  const int tg = (NG + 15) >> 4;
  const int td = (ND + 15) >> 4;

  // degrees -> reciprocal degrees (shared by both layers)
  (void)hipMemsetAsync(cnt_d, 0, (size_t)(ND + NG) * sizeof(float), stream);
  deg_kernel<<<2048, 256, 0, stream>>>(src, dst, cnt_d, cnt_g, E);
  inv_kernel<<<(ND + NG + 255) / 256, 256, 0, stream>>>(cnt_d, ND + NG);

  // ---- layer 1 ----
  (void)hipMemsetAsync(sum_g, 0, (size_t)NG * HD * sizeof(float), stream);
  (void)hipMemsetAsync(sum_d, 0, (size_t)ND * HD * sizeof(float), stream);
  scatter_sum_kernel<<<16384, 256, 0, stream>>>(x_d, src, dst, sum_g, E);  // gene <- disease
  scatter_sum_kernel<<<16384, 256, 0, stream>>>(x_g, dst, src, sum_d, E);  // disease <- gene

  sage_gemm_kernel<<<imin(tg, 1024), 256, 0, stream>>>(sum_g, cnt_g, x_g,
                                                       w1dgl, w1dgr, b1dg, g1, NG);
  sage_gemm_kernel<<<imin(td, 1024), 256, 0, stream>>>(sum_d, cnt_d, x_d,
                                                       w1gdl, w1gdr, b1gd, d1, ND);

  // ---- layer 2 ----
  (void)hipMemsetAsync(sum_g, 0, (size_t)NG * HD * sizeof(float), stream);
  (void)hipMemsetAsync(sum_d, 0, (size_t)ND * HD * sizeof(float), stream);
  scatter_sum_kernel<<<16384, 256, 0, stream>>>(d1, src, dst, sum_g, E);
  scatter_sum_kernel<<<16384, 256, 0, stream>>>(g1, dst, src, sum_d, E);

  sage_gemm_kernel<<<imin(tg, 1024), 256, 0, stream>>>(sum_g, cnt_g, g1,
                                                       w2dgl, w2dgr, b2dg, g2, NG);
  sage_gemm_kernel<<<imin(td, 1024), 256, 0, stream>>>(sum_d, cnt_d, d1,
                                                       w2gdl, w2gdr, b2gd, d2, ND);
}